// Yolov1_Loss_69870527971485
// MI455X (gfx1250) — compile-verified
//
#include <hip/hip_runtime.h>
#include <hip/hip_bf16.h>
#include <math.h>

// YOLOv1-style loss, fused gather + reduction for gfx1250 (wave32).
// pred: (B, 30, 7, 7) f32 ; annot: (B, 2, 5) f32 ; out: scalar f32.

#define SS    7
#define NBOX  2
#define NCLS  20
#define INV_IMG (1.0f / 448.0f)

typedef __attribute__((ext_vector_type(2))) float v2f;
typedef __attribute__((ext_vector_type(8))) float v8f;

__device__ __forceinline__ float sigmoidf_(float x) {
    return 1.0f / (1.0f + expf(-x));
}

// IoU with boxes as [x, y, w, h]
__device__ __forceinline__ float iou_f(const float* a, const float* g) {
    float x1 = fmaxf(a[0], g[0]);
    float y1 = fmaxf(a[1], g[1]);
    float x2 = fminf(a[0] + a[2], g[0] + g[2]);
    float y2 = fminf(a[1] + a[3], g[1] + g[3]);
    float inter = fmaxf(x2 - x1, 0.0f) * fmaxf(y2 - y1, 0.0f);
    float uni = a[2] * a[3] + g[2] * g[3] - inter;
    return inter / (uni + 1e-6f);
}

// Stage 1: one thread per batch item; fixed-order LDS tree reduce per block.
__global__ void __launch_bounds__(256)
yolo_loss_stage1(const float* __restrict__ pred,
                 const float* __restrict__ annot,
                 float* __restrict__ partial, int n) {
    const int tid = threadIdx.x;
    const int b = blockIdx.x * 256 + tid;

    float loss = 0.0f;
    if (b < n) {
        // ---- load annot (2 boxes x 5) ----
        float gt[NBOX][5];
        const float* ab = annot + (size_t)b * (NBOX * 5);
        #pragma unroll
        for (int k = 0; k < NBOX; ++k)
            #pragma unroll
            for (int j = 0; j < 5; ++j)
                gt[k][j] = ab[k * 5 + j];

        // ---- target cell indices ----
        int ti[NBOX], tj[NBOX];
        #pragma unroll
        for (int k = 0; k < NBOX; ++k) {
            float nx = gt[k][0] * INV_IMG;
            float ny = gt[k][1] * INV_IMG;
            float nw = gt[k][2] * INV_IMG;
            float nh = gt[k][3] * INV_IMG;
            ti[k] = (int)floorf((nx + nw * 0.5f) * (float)SS);
            tj[k] = (int)floorf((ny + nh * 0.5f) * (float)SS);
        }
        // clamp purely for memory safety (in-distribution they are already 0..6)
        int i1 = min(max(ti[1], 0), SS - 1);
        int j1 = min(max(tj[1], 0), SS - 1);

        // ---- gather 30 channels of the chosen cell (stride 49 dwords) ----
        const float* pb = pred + (size_t)b * (30 * SS * SS) + i1 * SS + j1;
        float raw[30];
        #pragma unroll
        for (int c = 0; c < 30; ++c) raw[c] = pb[c * (SS * SS)];

        float cell[30];
        #pragma unroll
        for (int c = 0; c < 30; ++c) cell[c] = sigmoidf_(raw[c]);

        const float* box1 = &cell[0];
        const float* box2 = &cell[5];
        const float* clsp = &cell[10];

        // ---- per-gt-box loss terms ----
        #pragma unroll
        for (int k = 0; k < NBOX; ++k) {
            float v1 = iou_f(box1, gt[k]);
            float v2 = iou_f(box2, gt[k]);
            const float* sel = (v1 > v2) ? box1 : box2;

            float g0 = gt[k][0] * INV_IMG;
            float g1 = gt[k][1] * INV_IMG;
            float g2 = gt[k][2] * INV_IMG;
            float g3 = gt[k][3] * INV_IMG;

            float dx = g0 - sel[0];
            float dy = g1 - sel[1];
            float xy = 5.0f * (dx * dx + dy * dy);

            float dw = sqrtf(g2) - sqrtf(sel[2]);
            float dh = sqrtf(g3) - sqrtf(sel[3]);
            float wh = 5.0f * (dw * dw + dh * dh);

            float ob = 1.0f - sel[4];
            float obj = ob * ob;

            int cc = (int)gt[k][4];
            float cls_loss = 0.0f;
            #pragma unroll
            for (int c = 0; c < NCLS; ++c) {
                float d = ((c == cc) ? 1.0f : 0.0f) - clsp[c];
                cls_loss = fmaf(d, d, cls_loss);
            }
            loss += xy + wh + obj + cls_loss;
        }

        // ---- no-object term: find last cell c=i*7+j with
        //      any_k ((i != ti[k]) && (j != tj[k])) ----
        int last = SS * SS - 1;  // argmax over all-false reversed mask -> 48
        for (int c = SS * SS - 1; c >= 0; --c) {
            int i = c / SS, j = c % SS;
            bool m = false;
            #pragma unroll
            for (int k = 0; k < NBOX; ++k)
                m = m || ((i != ti[k]) && (j != tj[k]));
            if (m) { last = c; break; }
        }
        float conf = sigmoidf_(pred[(size_t)b * (30 * SS * SS) + 9 * SS * SS + last]);
        loss += 2.0f * conf * conf;  // nobj broadcast over NB=2 gt boxes
    }

    // ---- deterministic fixed-order block reduction ----
    __shared__ float sdata[256];
    sdata[tid] = loss;
    __syncthreads();
    #pragma unroll
    for (int s = 128; s > 0; s >>= 1) {
        if (tid < s) sdata[tid] += sdata[tid + s];
        __syncthreads();
    }
    if (tid == 0) partial[blockIdx.x] = sdata[0];
}

// Stage 2: single wave32. Reduce n partials with V_WMMA_F32_16X16X4_F32:
// D = A x Ones + C, 64 partials per iteration packed into the 16x4 f32 A
// operand per the CDNA5 A-matrix layout (lanes 0-15: K=0,1; lanes 16-31: K=2,3).
// D rows accumulate across iterations through C. Full f32, deterministic.
__global__ void __launch_bounds__(32)
yolo_loss_stage2(const float* __restrict__ partial, int n,
                 float* __restrict__ out) {
    const int lane = threadIdx.x;  // 0..31, EXEC all ones (required for WMMA)

    v8f acc = {};
    v2f ones;
    ones.x = 1.0f;
    ones.y = 1.0f;

    const int row = lane & 15;
    const int khi = (lane >> 4) << 1;  // 0 for lanes 0-15, 2 for lanes 16-31

    for (int base = 0; base < n; base += 64) {
        int idx0 = base + row * 4 + khi;
        v2f a;
        a.x = (idx0     < n) ? partial[idx0]     : 0.0f;
        a.y = (idx0 + 1 < n) ? partial[idx0 + 1] : 0.0f;
        // 8 args: (neg_a, A, neg_b, B, c_mod, C, reuse_a, reuse_b)
        acc = __builtin_amdgcn_wmma_f32_16x16x4_f32(
            false, a, false, ones, (short)0, acc, false, false);
    }

    // B == ones so every column of D is identical.
    // Lane 0 holds D[0..7][0] across its 8 D VGPRs, lane 16 holds D[8..15][0].
    float s = acc[0] + acc[1] + acc[2] + acc[3] +
              acc[4] + acc[5] + acc[6] + acc[7];

    __shared__ float red[2];
    if (lane == 0)  red[0] = s;
    if (lane == 16) red[1] = s;
    __syncthreads();
    if (lane == 0) out[0] = red[0] + red[1];
}

extern "C" void kernel_launch(void* const* d_in, const int* in_sizes, int n_in,
                              void* d_out, int out_size, void* d_ws, size_t ws_size,
                              hipStream_t stream) {
    const float* pred  = (const float*)d_in[0];
    const float* annot = (const float*)d_in[1];
    float* out = (float*)d_out;

    int n = in_sizes[1] / (NBOX * 5);   // batch size (16384)
    int nblk = (n + 255) / 256;         // 64 partials

    float* part = (float*)d_ws;         // nblk * 4 bytes of scratch

    yolo_loss_stage1<<<nblk, 256, 0, stream>>>(pred, annot, part, n);
    yolo_loss_stage2<<<1, 32, 0, stream>>>(part, nblk, out);
}